// ConceptGNN_53085795779125
// MI455X (gfx1250) — compile-verified
//
#include <hip/hip_runtime.h>

// Problem constants (match reference)
#define Vn 50000
#define Dn 300
#define Gn 256
#define En 1600000
#define Bn 64
#define Sn 64

typedef __attribute__((ext_vector_type(2))) float v2f;
typedef __attribute__((ext_vector_type(8))) float v8f;

// ---------------- degree / normalization ----------------
__global__ void k_init_deg(float* __restrict__ deg) {
  int i = blockIdx.x * blockDim.x + threadIdx.x;
  if (i < Vn) deg[i] = 1.0f;  // self-loop
}

__global__ void k_deg_accum(float* __restrict__ deg, const int* __restrict__ dst) {
  int e = blockIdx.x * blockDim.x + threadIdx.x;
  if (e < En) atomicAdd(&deg[dst[e]], 1.0f);
}

__global__ void k_rsqrt(float* __restrict__ deg) {
  int i = blockIdx.x * blockDim.x + threadIdx.x;
  if (i < Vn) deg[i] = rsqrtf(deg[i]);
}

// ---------------- GEMM: H[V,256] = X[V,K] @ W[K,256] via V_WMMA_F32_16X16X4_F32 ----------------
// blockDim = 256 (8 waves), grid = V/80 = 625 blocks.
// Each wave owns an 80x32 register tile: 5 M-subtiles x 2 N-subtiles = 10 v8f accumulators
// (80 accumulator VGPRs -- fits without spilling, unlike the 5x4 tile).
// Per k-step: 5 A-frag loads (b64) + 4 B-frag loads (b32) feed 10 WMMAs.
// V = 625*80 and G = 8*32 exactly -> no partial tiles, EXEC all-1s for WMMA.
__global__ void __launch_bounds__(256, 1)
k_gemm_wmma(const float* __restrict__ X, const float* __restrict__ W,
            float* __restrict__ H, int K) {
  const int wave = threadIdx.x >> 5;      // 0..7 -> N-group of 32 columns
  const int lane = threadIdx.x & 31;
  const int l15  = lane & 15;
  const int hi   = lane >> 4;             // lane-half selects K sub-slice
  const int rbase = blockIdx.x * 80;      // first row of this block's 5 M-subtiles
  const int ncol  = wave * 32 + l15;      // B/C column (N = lane&15)

  v8f acc[5][2] = {};

  const float* xp = X + (size_t)(rbase + l15) * K;  // lane's row in M-subtile 0

  for (int k = 0; k < K; k += 4) {
    const int ka = k + hi * 2;  // ISA A layout: lanes 0-15 hold K=k,k+1; lanes 16-31 hold K=k+2,k+3
    v2f a[5];
#pragma unroll
    for (int m = 0; m < 5; ++m) {
      const float* p = xp + (size_t)(m * 16) * K + ka;
      a[m].x = p[0];
      a[m].y = p[1];
    }

    // ISA B layout (4x16 f32): VGPR0 = row K=ka, VGPR1 = row K=ka+1, N = lane&15 per half-wave
    const float* w0 = W + (size_t)ka * Gn;
    const float* w1 = w0 + Gn;
    v2f b[2];
#pragma unroll
    for (int n = 0; n < 2; ++n) {
      b[n].x = w0[ncol + 16 * n];
      b[n].y = w1[ncol + 16 * n];
    }

#pragma unroll
    for (int m = 0; m < 5; ++m)
#pragma unroll
      for (int n = 0; n < 2; ++n)
        acc[m][n] = __builtin_amdgcn_wmma_f32_16x16x4_f32(
            false, a[m], false, b[n], (short)0, acc[m][n], false, false);
  }

  // C/D layout: VGPR r -> M = r + 8*hi, N = lane&15
#pragma unroll
  for (int m = 0; m < 5; ++m) {
    const int srow = rbase + m * 16 + hi * 8;
    float* __restrict__ hp = H + (size_t)srow * Gn + ncol;
#pragma unroll
    for (int r = 0; r < 8; ++r) {
      hp[(size_t)r * Gn +  0] = acc[m][0][r];
      hp[(size_t)r * Gn + 16] = acc[m][1][r];
    }
  }
}

// ---------------- self-loop init: B = A * inv[i]^2 + bias ----------------
__global__ void k_selfloop(float* __restrict__ Bm, const float* __restrict__ A,
                           const float* __restrict__ inv, const float* __restrict__ bias) {
  int t = blockIdx.x * blockDim.x + threadIdx.x;   // V*64 threads, float4 each
  if (t >= Vn * 64) return;
  int i = t >> 6;
  int c = (t & 63) << 2;
  float s = inv[i] * inv[i];
  const float4 v = *(const float4*)(A + (size_t)i * Gn + c);
  const float4 b = *(const float4*)(bias + c);
  float4 o;
  o.x = v.x * s + b.x; o.y = v.y * s + b.y;
  o.z = v.z * s + b.z; o.w = v.w * s + b.w;
  *(float4*)(Bm + (size_t)i * Gn + c) = o;
}

// ---------------- edge scatter: B[dst] += A[src] * inv[src]*inv[dst] ----------------
__global__ void k_scatter(float* __restrict__ Bm, const float* __restrict__ A,
                          const float* __restrict__ inv, const int* __restrict__ src,
                          const int* __restrict__ dst) {
  long long t = (long long)blockIdx.x * blockDim.x + threadIdx.x;  // E*64 threads
  int e = (int)(t >> 6);
  if (e >= En) return;
  int c = ((int)t & 63) << 2;
  int s = src[e], d = dst[e];
  float coef = inv[s] * inv[d];
  const float4 v = *(const float4*)(A + (size_t)s * Gn + c);
  float* p = Bm + (size_t)d * Gn + c;
  atomicAdd(p + 0, v.x * coef);
  atomicAdd(p + 1, v.y * coef);
  atomicAdd(p + 2, v.z * coef);
  atomicAdd(p + 3, v.w * coef);
}

// ---------------- relu in place ----------------
__global__ void k_relu(float* __restrict__ Bm) {
  int t = blockIdx.x * blockDim.x + threadIdx.x;   // V*64 threads, float4 each
  if (t >= Vn * 64) return;
  float4 v = *(float4*)(Bm + (size_t)t * 4);
  v.x = v.x > 0.f ? v.x : 0.f;
  v.y = v.y > 0.f ? v.y : 0.f;
  v.z = v.z > 0.f ? v.z : 0.f;
  v.w = v.w > 0.f ? v.w : 0.f;
  *(float4*)(Bm + (size_t)t * 4) = v;
}

// ---------------- gather rows: out[r,:] = B[word_ids[r],:] ----------------
__global__ void k_gather(float* __restrict__ out, const float* __restrict__ Bm,
                         const int* __restrict__ ids) {
  int r = blockIdx.x;            // 0..4095
  int c = threadIdx.x << 2;      // 64 threads * float4 = 256
  int id = ids[r];
  *(float4*)(out + (size_t)r * Gn + c) = *(const float4*)(Bm + (size_t)id * Gn + c);
}

extern "C" void kernel_launch(void* const* d_in, const int* in_sizes, int n_in,
                              void* d_out, int out_size, void* d_ws, size_t ws_size,
                              hipStream_t stream) {
  const float* emb = (const float*)d_in[0];
  const float* W1  = (const float*)d_in[1];
  const float* b1  = (const float*)d_in[2];
  const float* W2  = (const float*)d_in[3];
  const float* b2  = (const float*)d_in[4];
  const int*   ei  = (const int*)d_in[5];     // [2, E]: row 0 = src, row 1 = dst
  const int*   wid = (const int*)d_in[6];     // [B, S]
  float* out = (float*)d_out;

  const int* src = ei;
  const int* dst = ei + En;

  // Workspace: deg/inv (V, padded) | A (V*G) | B (V*G)  ~= 103 MB
  float* deg = (float*)d_ws;
  float* A   = deg + ((Vn + 255) & ~255);     // h1, later h2
  float* Bm  = A + (size_t)Vn * Gn;           // agg1, later agg2

  const int T = 256;
  // degree -> inv_sqrt
  k_init_deg <<<(Vn + T - 1) / T, T, 0, stream>>>(deg);
  k_deg_accum<<<(En + T - 1) / T, T, 0, stream>>>(deg, dst);
  k_rsqrt    <<<(Vn + T - 1) / T, T, 0, stream>>>(deg);

  // ---- layer 1 ----
  k_gemm_wmma<<<Vn / 80, 256, 0, stream>>>(emb, W1, A, Dn);             // A = emb @ W1
  k_selfloop <<<(Vn * 64 + T - 1) / T, T, 0, stream>>>(Bm, A, deg, b1); // B = A*inv^2 + b1
  k_scatter  <<<(int)(((long long)En * 64 + T - 1) / T), T, 0, stream>>>(Bm, A, deg, src, dst);
  k_relu     <<<(Vn * 64 + T - 1) / T, T, 0, stream>>>(Bm);

  // ---- layer 2 ----
  k_gemm_wmma<<<Vn / 80, 256, 0, stream>>>(Bm, W2, A, Gn);              // A = relu(agg1) @ W2
  k_selfloop <<<(Vn * 64 + T - 1) / T, T, 0, stream>>>(Bm, A, deg, b2); // B = A*inv^2 + b2 (agg1 dead after GEMM2)
  k_scatter  <<<(int)(((long long)En * 64 + T - 1) / T), T, 0, stream>>>(Bm, A, deg, src, dst);

  // ---- gather output ----
  k_gather<<<Bn * Sn, 64, 0, stream>>>(out, Bm, wid);

  (void)in_sizes; (void)n_in; (void)out_size; (void)ws_size;
}